// GHM_Loss_5669356834621
// MI455X (gfx1250) — compile-verified
//
#include <hip/hip_runtime.h>
#include <hip/hip_bf16.h>

#define GHM_BINS 10
#define GHM_EPS  1e-4f

// d_ws layout (32-bit words): [0..9] u32 bin counts, [16..25] f32 bin bce-sums.

__global__ void ghm_init_ws(unsigned* __restrict__ ws) {
    int t = threadIdx.x;
    if (t < 32) ws[t] = 0u;   // zeros both the u32 counts and the f32 sums (0.0f)
}

__global__ __launch_bounds__(256) void ghm_main(const float* __restrict__ x,
                                                const float* __restrict__ tgt,
                                                unsigned* __restrict__ ws,
                                                int nv4 /* N/4 */) {
    // Per-wave double-buffered staging: 8 waves x 2 buffers x 32 lanes x 16B, per array.
    __shared__ float4 stg_x[8][2][32];
    __shared__ float4 stg_t[8][2][32];
    // Half-wave replicated histograms (padded 10 -> 16 bins).
    __shared__ unsigned h_cnt[16][16];
    __shared__ float    h_sum[16][16];

    const int tid  = threadIdx.x;
    const int lane = tid & 31;
    const int wid  = tid >> 5;   // wave id in block (wave32)
    const int rep  = tid >> 4;   // half-wave replica id 0..15

    h_cnt[tid >> 4][tid & 15] = 0u;
    h_sum[tid >> 4][tid & 15] = 0.0f;
    __syncthreads();

    const int waves_total = gridDim.x * (blockDim.x >> 5);
    const int gwave       = blockIdx.x * (blockDim.x >> 5) + wid;
    const int nchunks     = nv4 >> 5;          // chunk = 32 consecutive float4 per array

    // Issue one chunk of async copies (global -> LDS), 16B per lane per array.
    auto issue = [&](int chunk, int buf) {
        const float4* gx = (const float4*)x   + ((size_t)chunk * 32 + lane);
        const float4* gt = (const float4*)tgt + ((size_t)chunk * 32 + lane);
        unsigned lx = (unsigned)(size_t)&stg_x[wid][buf][lane];  // low 32 bits = LDS offset
        unsigned lt = (unsigned)(size_t)&stg_t[wid][buf][lane];
        unsigned long long ax = (unsigned long long)gx;
        unsigned long long at = (unsigned long long)gt;
        asm volatile("global_load_async_to_lds_b128 %0, %1, off"
                     :: "v"(lx), "v"(ax) : "memory");
        asm volatile("global_load_async_to_lds_b128 %0, %1, off"
                     :: "v"(lt), "v"(at) : "memory");
    };

    auto proc = [&](float xs, float ts) {
        float ax  = fabsf(xs);
        float e   = __builtin_amdgcn_exp2f(-1.44269504088896340736f * ax); // exp(-|x|)
        float d   = 1.0f + e;
        float r   = __builtin_amdgcn_rcpf(d);
        float sig = (xs >= 0.0f) ? r : e * r;             // sigmoid(x)
        float g   = fabsf(sig - ts);
        int bin   = (int)(g * (10.0f - 1e-4f));           // trunc == floor for g >= 0
        bin       = bin > 9 ? 9 : bin;
        float bce = fmaxf(xs, 0.0f) - xs * ts
                  + 0.69314718055994530942f * __builtin_amdgcn_logf(d);   // + log1p(e)
        atomicAdd(&h_cnt[rep][bin], 1u);                  // ds_add_u32 (exact count)
        atomicAdd(&h_sum[rep][bin], bce);                 // ds_add_f32
    };

    int c = gwave;
    int p = 0;
    if (c < nchunks) issue(c, 0);
    while (c < nchunks) {
        int cn = c + waves_total;
        if (cn < nchunks) {
            issue(cn, p ^ 1);
            asm volatile("s_wait_asynccnt 2" ::: "memory"); // current chunk landed
        } else {
            asm volatile("s_wait_asynccnt 0" ::: "memory");
        }
        float4 xv = stg_x[wid][p][lane];   // ds_load_b128
        float4 tv = stg_t[wid][p][lane];
        proc(xv.x, tv.x);
        proc(xv.y, tv.y);
        proc(xv.z, tv.z);
        proc(xv.w, tv.w);
        c = cn;
        p ^= 1;
    }

    __syncthreads();
    // Flush block-local histogram to global accumulators.
    if (tid < GHM_BINS) {
        unsigned s = 0u;
        #pragma unroll
        for (int r2 = 0; r2 < 16; ++r2) s += h_cnt[r2][tid];
        atomicAdd(&ws[tid], s);
    } else if (tid >= 32 && tid < 32 + GHM_BINS) {
        int b = tid - 32;
        float s = 0.0f;
        #pragma unroll
        for (int r2 = 0; r2 < 16; ++r2) s += h_sum[r2][b];
        atomicAdd((float*)ws + 16 + b, s);
    }
}

__global__ void ghm_final(const unsigned* __restrict__ ws, float* __restrict__ out) {
    if (threadIdx.x == 0) {
        const unsigned* cnt = ws;
        const float*    sum = (const float*)(ws + 16);
        float ne = 0.0f;
        #pragma unroll
        for (int b = 0; b < GHM_BINS; ++b) ne += (cnt[b] > 0u) ? 1.0f : 0.0f;
        float acc = 0.0f;
        #pragma unroll
        for (int b = 0; b < GHM_BINS; ++b) {
            float gd = fmaxf((float)cnt[b] * ne, GHM_EPS);
            acc += sum[b] / gd;     // == (1/N) * sum_i beta[bin_i]*bce_i
        }
        out[0] = acc;
    }
}

extern "C" void kernel_launch(void* const* d_in, const int* in_sizes, int n_in,
                              void* d_out, int out_size, void* d_ws, size_t ws_size,
                              hipStream_t stream) {
    const float* x = (const float*)d_in[0];
    const float* t = (const float*)d_in[1];
    float*    out  = (float*)d_out;
    unsigned* ws   = (unsigned*)d_ws;

    long long n   = (long long)in_sizes[0];   // 8192*8192
    int       nv4 = (int)(n >> 2);

    ghm_init_ws<<<1, 32, 0, stream>>>(ws);
    ghm_main<<<2048, 256, 0, stream>>>(x, t, ws, nv4);
    ghm_final<<<1, 32, 0, stream>>>(ws, out);
}